// PatialEERCRF_14559939133968
// MI455X (gfx1250) — compile-verified
//
#include <hip/hip_runtime.h>
#include <hip/hip_bf16.h>

// ---------------------------------------------------------------------------
// CRF loss via chunked parallel scan in the exp domain, using f32 WMMA
// (V_WMMA_F32_16X16X4_F32) for the per-step 32x32 operator products.
// ---------------------------------------------------------------------------

#define CC 25          // number of tags (1 + 2*12)
#define PP 32          // padded to 32 for WMMA tiles
#define BB 64          // batch
#define NN 1024        // sequence length
#define LCH 64         // chunk length (steps per chunk)
#define NCH 16         // ceil((NN-1)/LCH) = ceil(1023/64)

typedef __attribute__((ext_vector_type(2))) float v2f;
typedef __attribute__((ext_vector_type(8))) float v8f;

__device__ __forceinline__ float wave_max32(float v) {
  #pragma unroll
  for (int off = 16; off > 0; off >>= 1) v = fmaxf(v, __shfl_xor(v, off, 32));
  return v;
}
__device__ __forceinline__ float wave_sum32(float v) {
  #pragma unroll
  for (int off = 16; off > 0; off >>= 1) v += __shfl_xor(v, off, 32);
  return v;
}

// ---------------------------------------------------------------------------
// Kernel 1: each wave computes the 32x32 product of its chunk's step
// operators  P_c = prod_t ( expT * diag(w_t) ),  t in [1+c*L, min(1+(c+1)*L,N))
// with per-step max-renormalization tracked in a scalar log-scale.
// grid = 2 passes * 64 batch * 16 chunks = 2048 single-wave blocks.
// ---------------------------------------------------------------------------
__global__ __launch_bounds__(32) void crf_chunk_ops(
    const float* __restrict__ tag_scores,   // (B,N,C)
    const float* __restrict__ trans,        // (C,C)
    const float* __restrict__ constr,       // (C,C)
    const int*   __restrict__ labels,       // (B,N)
    const int*   __restrict__ amask,        // (B,N)
    float* __restrict__ mats,               // (2*B*NCH, 32*32)
    float* __restrict__ lsc)                // (2*B*NCH)
{
  const int lane = threadIdx.x;
  const int bx   = blockIdx.x;
  const int c    = bx & (NCH - 1);
  const int b    = (bx >> 4) & (BB - 1);
  const int pass = bx >> 10;                // 0 = unconstrained, 1 = constrained

  __shared__ float ldsT[PP * PP];
  __shared__ float ldsM[PP * PP];

  // Build expT, padded 32x32 with zeros; disallowed transitions -> exact 0.
  for (int idx = lane; idx < PP * PP; idx += 32) {
    int r = idx >> 5, cl = idx & 31;
    float v = 0.0f;
    if (r < CC && cl < CC)
      v = (constr[r * CC + cl] > 0.5f) ? __expf(trans[r * CC + cl]) : 0.0f;
    ldsT[idx] = v;
  }
  __syncthreads();

  // expT resident in B-operand layout for the whole chunk.
  // B 4x16 f32 tile: V0 = {K=0 | lanes 0-15, K=2 | lanes 16-31}, V1 = {K=1, K=3}.
  const int n0    = lane & 15;
  const int khalf = (lane >> 4) << 1;       // 0 or 2
  const int mhalf = (lane >> 4) << 3;       // 0 or 8 (for C/D layout)
  v2f Tb[8][2];
  #pragma unroll
  for (int kc = 0; kc < 8; ++kc)
    #pragma unroll
    for (int nt = 0; nt < 2; ++nt) {
      int k = kc * 4 + khalf;
      int n = nt * 16 + n0;
      Tb[kc][nt].x = ldsT[k * PP + n];
      Tb[kc][nt].y = ldsT[(k + 1) * PP + n];
    }

  // M = identity in C/D layout (tile r holds rows r and r+8).
  v8f M[2][2];
  #pragma unroll
  for (int mt = 0; mt < 2; ++mt)
    #pragma unroll
    for (int nt = 0; nt < 2; ++nt)
      #pragma unroll
      for (int r = 0; r < 8; ++r)
        M[mt][nt][r] = ((mt * 16 + mhalf + r) == (nt * 16 + n0)) ? 1.0f : 0.0f;

  float ls = 0.0f;

  const int tbeg = 1 + c * LCH;
  int tend = tbeg + LCH; if (tend > NN) tend = NN;

  for (int t = tbeg; t < tend; ++t) {
    if (amask[b * NN + t] == 0) continue;   // masked step == identity operator

    // D-layout -> LDS (row major 32x32)
    #pragma unroll
    for (int mt = 0; mt < 2; ++mt)
      #pragma unroll
      for (int nt = 0; nt < 2; ++nt)
        #pragma unroll
        for (int r = 0; r < 8; ++r)
          ldsM[(mt * 16 + mhalf + r) * PP + nt * 16 + n0] = M[mt][nt][r];
    __syncthreads();

    // LDS -> A-operand layout (16x4 f32: lanes 0-15 carry K=0,1; 16-31 K=2,3)
    v2f A[2][8];
    #pragma unroll
    for (int mt = 0; mt < 2; ++mt)
      #pragma unroll
      for (int kc = 0; kc < 8; ++kc) {
        int m = mt * 16 + n0;
        int k = kc * 4 + khalf;
        A[mt][kc].x = ldsM[m * PP + k];
        A[mt][kc].y = ldsM[m * PP + k + 1];
      }
    __syncthreads();

    // Per-step diagonal weights w = exp(emission) (constrained pass: one-hot).
    const float* em = tag_scores + ((size_t)b * NN + t) * CC;
    const int lab = labels[b * NN + t];
    float wv[2];
    #pragma unroll
    for (int nt = 0; nt < 2; ++nt) {
      int col = nt * 16 + n0;
      float v = 0.0f;
      if (col < CC && (pass == 0 || col == lab)) v = __expf(em[col]);
      wv[nt] = v;
    }

    // G = M * expT  via 32 x V_WMMA_F32_16X16X4_F32, then column-scale by w.
    #pragma unroll
    for (int mt = 0; mt < 2; ++mt)
      #pragma unroll
      for (int nt = 0; nt < 2; ++nt) {
        v8f g = {};
        #pragma unroll
        for (int kc = 0; kc < 8; ++kc)
          g = __builtin_amdgcn_wmma_f32_16x16x4_f32(
                false, A[mt][kc], false, Tb[kc][nt],
                (short)0, g, false, false);
        #pragma unroll
        for (int r = 0; r < 8; ++r) g[r] *= wv[nt];
        M[mt][nt] = g;
      }

    // Renormalize by global max; fold into log-scale.
    float mx = 0.0f;
    #pragma unroll
    for (int mt = 0; mt < 2; ++mt)
      #pragma unroll
      for (int nt = 0; nt < 2; ++nt)
        #pragma unroll
        for (int r = 0; r < 8; ++r) mx = fmaxf(mx, M[mt][nt][r]);
    mx = fmaxf(wave_max32(mx), 1e-30f);
    const float inv = 1.0f / mx;
    ls += __logf(mx);
    #pragma unroll
    for (int mt = 0; mt < 2; ++mt)
      #pragma unroll
      for (int nt = 0; nt < 2; ++nt)
        #pragma unroll
        for (int r = 0; r < 8; ++r) M[mt][nt][r] *= inv;
  }

  // Store chunk operator (plain row-major) + its log-scale.
  const size_t base = (size_t)bx * (PP * PP);
  #pragma unroll
  for (int mt = 0; mt < 2; ++mt)
    #pragma unroll
    for (int nt = 0; nt < 2; ++nt)
      #pragma unroll
      for (int r = 0; r < 8; ++r)
        mats[base + (size_t)(mt * 16 + mhalf + r) * PP + nt * 16 + n0] = M[mt][nt][r];
  if (lane == 0) lsc[bx] = ls;
}

// ---------------------------------------------------------------------------
// Kernel 2: per (pass, batch) fold:  a <- a * P_c  over 16 chunks,
// accumulating log-scales; emit Z = ls + log(sum a).  128 single-wave blocks.
// ---------------------------------------------------------------------------
__global__ __launch_bounds__(32) void crf_fold(
    const float* __restrict__ tag_scores,
    const int*   __restrict__ labels,
    const float* __restrict__ mats,
    const float* __restrict__ lsc,
    float* __restrict__ Zbuf)               // (2*B)
{
  const int lane = threadIdx.x;
  const int b    = blockIdx.x & (BB - 1);
  const int pass = blockIdx.x >> 6;

  __shared__ float av[PP];

  // alpha0 = exp(emissions at t=0), constrained pass keeps only the label tag.
  float a = 0.0f;
  const int lab0 = labels[b * NN + 0];
  if (lane < CC) {
    float e0 = tag_scores[(size_t)b * NN * CC + lane];
    if (pass == 0 || lane == lab0) a = __expf(e0);
  }
  float mx = fmaxf(wave_max32(a), 1e-30f);
  a /= mx;
  float ls = __logf(mx);

  for (int c = 0; c < NCH; ++c) {
    const int task = (pass * BB + b) * NCH + c;
    const float* Pm = mats + (size_t)task * (PP * PP);
    av[lane] = a;
    __syncthreads();
    float s = 0.0f;
    #pragma unroll 8
    for (int k = 0; k < PP; ++k)
      s = fmaf(av[k], Pm[k * PP + lane], s);
    __syncthreads();
    a = s;
    ls += lsc[task];
    float m2 = fmaxf(wave_max32(a), 1e-30f);
    a /= m2;
    ls += __logf(m2);
  }

  const float tot = wave_sum32(a);
  if (lane == 0) Zbuf[pass * BB + b] = ls + __logf(tot);
}

// ---------------------------------------------------------------------------
// Kernel 3: loss = mean_b(Z - Zc) / 10
// ---------------------------------------------------------------------------
__global__ __launch_bounds__(64) void crf_loss(const float* __restrict__ Zbuf,
                                               float* __restrict__ out)
{
  __shared__ float red[64];
  const int t = threadIdx.x;
  red[t] = Zbuf[t] - Zbuf[BB + t];
  __syncthreads();
  #pragma unroll
  for (int off = 32; off > 0; off >>= 1) {
    if (t < off) red[t] += red[t + off];
    __syncthreads();
  }
  if (t == 0) out[0] = red[0] * (1.0f / (BB * 10.0f));
}

// ---------------------------------------------------------------------------
extern "C" void kernel_launch(void* const* d_in, const int* in_sizes, int n_in,
                              void* d_out, int out_size, void* d_ws, size_t ws_size,
                              hipStream_t stream) {
  const float* tag_scores = (const float*)d_in[0];   // (64,1024,25) f32
  const float* trans      = (const float*)d_in[1];   // (25,25)      f32
  const float* constr     = (const float*)d_in[2];   // (25,25)      f32
  const int*   labels     = (const int*)d_in[3];     // (64,1024)    int
  const int*   amask      = (const int*)d_in[4];     // (64,1024)    int

  const int ntask = 2 * BB * NCH;                    // 2048
  float* mats = (float*)d_ws;                        // ntask * 1024 f32 (8 MB)
  float* lsc  = mats + (size_t)ntask * (PP * PP);    // ntask f32
  float* Zbuf = lsc + ntask;                         // 2*BB f32

  crf_chunk_ops<<<dim3(ntask), dim3(32), 0, stream>>>(
      tag_scores, trans, constr, labels, amask, mats, lsc);
  crf_fold<<<dim3(2 * BB), dim3(32), 0, stream>>>(
      tag_scores, labels, mats, lsc, Zbuf);
  crf_loss<<<dim3(1), dim3(64), 0, stream>>>(Zbuf, (float*)d_out);
}